// CutMix_12232066859295
// MI455X (gfx1250) — compile-verified
//
#include <hip/hip_runtime.h>

// CutMix on MI455X (gfx1250): pure streaming select, HBM-bound (~450 MB @ 23.3 TB/s ≈ 20 us).
// Layout [B,H,W,C] = [64,512,512,3] f32. One wave32-aligned block per (b,h) row:
// row = 1536 floats = 384 float4 -> 384 threads (12 waves), one b128 load/store per lane.

#define B_ 64
#define H_ 512
#define W_ 512
#define C_ 3
#define ROW_FLOATS (W_ * C_)       // 1536
#define ROW_VEC4   (ROW_FLOATS/4)  // 384 threads per block

typedef float v4f __attribute__((ext_vector_type(4)));

__global__ __launch_bounds__(ROW_VEC4)
void CutMix_12232066859295_kernel(const float* __restrict__ x,
                                  const int*   __restrict__ y1,
                                  const int*   __restrict__ y2,
                                  const int*   __restrict__ x1,
                                  const int*   __restrict__ x2,
                                  const int*   __restrict__ perm,
                                  float*       __restrict__ out)
{
    // gfx1250 marker / no-op: TENSORcnt is 0 in this kernel (no TDM in flight).
    __builtin_amdgcn_s_wait_tensorcnt(0);

    const int row = blockIdx.x;          // row = b*H + h
    const int b   = row >> 9;            // H == 512
    const int h   = row & (H_ - 1);
    const int t   = threadIdx.x;         // 0..383, float4 index within row

    const long long rowOff = (long long)row * ROW_FLOATS;
    const v4f* __restrict__ xrow = (const v4f*)(x + rowOff);

    // 128-bit coalesced load of this row's float4 (default RT hint: keep x in
    // the 192MB L2 -- the shuffled gather re-reads parts of x).
    v4f v = xrow[t];

    // Block-uniform per-sample box params -> scalar loads + scalar branch.
    const int yb1 = y1[b];
    const int yb2 = y2[b];
    if (h >= yb1 && h < yb2) {
        const int lo   = x1[b] * C_;     // inclusive float index within row
        const int hi   = x2[b] * C_;     // exclusive
        const int base = t * 4;
        // Touch the shuffled image only where this float4 overlaps the box.
        if (base < hi && base + 4 > lo) {
            const int pb = perm[b];
            const v4f* __restrict__ srow =
                (const v4f*)(x + ((long long)pb * H_ + h) * ROW_FLOATS);
            v4f s = srow[t];
            // Interior float4s collapse to a straight copy; only the (<=2)
            // boundary float4s per row keep the per-component selects.
            v.x = (base + 0 >= lo && base + 0 < hi) ? s.x : v.x;
            v.y = (base + 1 >= lo && base + 1 < hi) ? s.y : v.y;
            v.z = (base + 2 >= lo && base + 2 < hi) ? s.z : v.z;
            v.w = (base + 3 >= lo && base + 3 < hi) ? s.w : v.w;
        }
    }

    // Output is write-once streaming: non-temporal b128 store so the result
    // does not evict x from L2.
    __builtin_nontemporal_store(v, (v4f*)(out + rowOff) + t);
}

extern "C" void kernel_launch(void* const* d_in, const int* in_sizes, int n_in,
                              void* d_out, int out_size, void* d_ws, size_t ws_size,
                              hipStream_t stream)
{
    (void)in_sizes; (void)n_in; (void)out_size; (void)d_ws; (void)ws_size;

    const float* x    = (const float*)d_in[0];
    const int*   y1   = (const int*)  d_in[1];
    const int*   y2   = (const int*)  d_in[2];
    const int*   x1   = (const int*)  d_in[3];
    const int*   x2   = (const int*)  d_in[4];
    const int*   perm = (const int*)  d_in[5];
    float*       out  = (float*)d_out;

    dim3 grid(B_ * H_);      // 32768 rows
    dim3 block(ROW_VEC4);    // 384 threads = 12 wave32s
    CutMix_12232066859295_kernel<<<grid, block, 0, stream>>>(
        x, y1, y2, x1, x2, perm, out);
}